// BatchIndexedMLP_59425167507528
// MI455X (gfx1250) — compile-verified
//
#include <hip/hip_runtime.h>

// BatchIndexedMLP fused kernel for gfx1250 (MI455X).
//
// Reference collapses to:  scatter-add -> GEMM+ELU -> GEMM+ELU -> GEMM -> gather-sum
// Everything fused in ONE kernel: 32 workgroups x 256 threads (8 wave32s),
// each WG owns 32 batch rows; XE/H/H2/T tiles live in LDS; GEMMs use
// V_WMMA_F32_16X16X4_F32 (fp32 matrix op => reference precision preserved).

typedef float v2f __attribute__((ext_vector_type(2)));
typedef float v8f __attribute__((ext_vector_type(8)));

#define NIN   126
#define NOUT  126
#define H1    256
#define H2D   256
#define KIDX  3
#define BT    32          // batch rows per workgroup
#define XES   132         // padded LDS stride for 128-wide tiles (bank-conflict-free A frags)
#define HS    260         // padded LDS stride for 256-wide tiles

__device__ __forceinline__ float elu_f(float v) {
    return v > 0.0f ? v : (__expf(v) - 1.0f);
}

__global__ __launch_bounds__(256, 1)
void fused_batch_indexed_mlp(const float* __restrict__ x,
                             const float* __restrict__ W0, const float* __restrict__ b0,
                             const float* __restrict__ W1, const float* __restrict__ b1,
                             const float* __restrict__ W2, const float* __restrict__ b2,
                             const int*   __restrict__ bii, const int* __restrict__ boi,
                             float* __restrict__ out)
{
    __shared__ float sCopy[KIDX][BT][128];  // deterministic scatter: one copy per k-slot
    __shared__ float sXE[BT][XES];          // XE tile, K padded to 128 (col 126 = pad row, col 127 = 0)
    __shared__ float sH [BT][HS];           // layer-0 activations
    __shared__ float sH2[BT][HS];           // layer-1 activations
    __shared__ float sT [BT][XES];          // layer-2 pre-gather output (cols 126/127 forced to 0)

    const int tid   = threadIdx.x;
    const int lane  = tid & 31;
    const int wave  = tid >> 5;
    const int m0    = lane & 15;            // row/col within 16x16 tile
    const int koff  = (lane >> 4) << 1;     // fp32 WMMA half-wave K offset (0 or 2)
    const int rbase = (lane >> 4) << 3;     // C/D fragment row base (0 or 8)
    const int brow0 = blockIdx.x * BT;

    // ---- warm L2 with the weights while the scatter phase runs (global_prefetch_b8) ----
    for (int i = tid * 32; i < H1 * NIN; i += 256 * 32)  __builtin_prefetch(&W0[i], 0, 1);
    for (int i = tid * 32; i < H2D * H1; i += 256 * 32)  __builtin_prefetch(&W1[i], 0, 1);
    for (int i = tid * 32; i < NOUT * H2D; i += 256 * 32) __builtin_prefetch(&W2[i], 0, 1);

    // ================= phase 1: deterministic scatter-add  XE[b,j] =================
    for (int i = tid; i < KIDX * BT * 128; i += 256)
        (&sCopy[0][0][0])[i] = 0.0f;
    __syncthreads();

    if (tid < BT * KIDX) {                  // 96 active threads: (row, k-slot)
        const int r = tid / KIDX;
        const int k = tid - r * KIDX;
        const int b = brow0 + r;
        const float* xr  = x + (size_t)b * NIN;
        const int*   ip  = bii + ((size_t)b * NIN) * KIDX + k;
        float* dst = sCopy[k][r];
        for (int i = 0; i < NIN; ++i)
            dst[ip[i * KIDX]] += xr[i];     // fixed order per (row,k) => deterministic
    }
    __syncthreads();

    for (int e = tid; e < BT * 128; e += 256) {
        const int r = e >> 7, j = e & 127;
        sXE[r][j] = sCopy[0][r][j] + sCopy[1][r][j] + sCopy[2][r][j];
    }
    __syncthreads();

    // ================= phase 2: H = ELU(XE @ W0^T + b0)   (K=128 padded) =================
    {
        const int n0 = wave * 32;           // each wave: 32 cols x 32 rows = 4 16x16 tiles
        v8f acc[2][2] = {};
        #pragma unroll 4
        for (int kk = 0; kk < 128; kk += 4) {
            v2f a[2], bf[2];
            #pragma unroll
            for (int mt = 0; mt < 2; ++mt)
                a[mt] = *(const v2f*)&sXE[mt * 16 + m0][kk + koff];
            #pragma unroll
            for (int nt = 0; nt < 2; ++nt) {
                const int n  = n0 + nt * 16 + m0;
                const int k0 = kk + koff;                       // even; valid pairs end at 124
                const int kc = (k0 < NIN - 1) ? k0 : 0;         // clamp: load always in-bounds
                v2f w = *(const v2f*)&W0[(size_t)n * NIN + kc];
                bf[nt].x = (k0 < NIN - 1) ? w.x : 0.0f;         // zero pad rows 126/127
                bf[nt].y = (k0 < NIN - 1) ? w.y : 0.0f;
            }
            #pragma unroll
            for (int mt = 0; mt < 2; ++mt)
                #pragma unroll
                for (int nt = 0; nt < 2; ++nt)
                    acc[mt][nt] = __builtin_amdgcn_wmma_f32_16x16x4_f32(
                        false, a[mt], false, bf[nt], (short)0, acc[mt][nt], false, false);
        }
        #pragma unroll
        for (int mt = 0; mt < 2; ++mt)
            #pragma unroll
            for (int nt = 0; nt < 2; ++nt) {
                const int col = n0 + nt * 16 + m0;
                const float bias = b0[col];
                #pragma unroll
                for (int r = 0; r < 8; ++r)
                    sH[mt * 16 + rbase + r][col] = elu_f(acc[mt][nt][r] + bias);
            }
    }
    __syncthreads();

    // ================= phase 3: H2 = ELU(H @ W1^T + b1)   (K=256) =================
    {
        const int n0 = wave * 32;
        v8f acc[2][2] = {};
        #pragma unroll 4
        for (int kk = 0; kk < H1; kk += 4) {
            v2f a[2], bf[2];
            #pragma unroll
            for (int mt = 0; mt < 2; ++mt)
                a[mt] = *(const v2f*)&sH[mt * 16 + m0][kk + koff];
            #pragma unroll
            for (int nt = 0; nt < 2; ++nt) {
                const int n = n0 + nt * 16 + m0;
                bf[nt] = *(const v2f*)&W1[(size_t)n * H1 + kk + koff];
            }
            #pragma unroll
            for (int mt = 0; mt < 2; ++mt)
                #pragma unroll
                for (int nt = 0; nt < 2; ++nt)
                    acc[mt][nt] = __builtin_amdgcn_wmma_f32_16x16x4_f32(
                        false, a[mt], false, bf[nt], (short)0, acc[mt][nt], false, false);
        }
        #pragma unroll
        for (int mt = 0; mt < 2; ++mt)
            #pragma unroll
            for (int nt = 0; nt < 2; ++nt) {
                const int col = n0 + nt * 16 + m0;
                const float bias = b1[col];
                #pragma unroll
                for (int r = 0; r < 8; ++r)
                    sH2[mt * 16 + rbase + r][col] = elu_f(acc[mt][nt][r] + bias);
            }
    }
    __syncthreads();

    // ================= phase 4: T = H2 @ W2^T + b2   (N padded to 128, cols>=126 -> 0) =================
    {
        const int n0 = wave * 16;           // each wave: 16 cols x 32 rows = 2 tiles
        const int n  = n0 + m0;
        const int nc = (n < NOUT) ? n : 0;
        const float nmask = (n < NOUT) ? 1.0f : 0.0f;
        v8f acc[2] = {};
        #pragma unroll 4
        for (int kk = 0; kk < H2D; kk += 4) {
            v2f a[2];
            #pragma unroll
            for (int mt = 0; mt < 2; ++mt)
                a[mt] = *(const v2f*)&sH2[mt * 16 + m0][kk + koff];
            v2f w = *(const v2f*)&W2[(size_t)nc * H2D + kk + koff];
            v2f bf; bf.x = w.x * nmask; bf.y = w.y * nmask;
            #pragma unroll
            for (int mt = 0; mt < 2; ++mt)
                acc[mt] = __builtin_amdgcn_wmma_f32_16x16x4_f32(
                    false, a[mt], false, bf, (short)0, acc[mt], false, false);
        }
        const float bias = ((n < NOUT) ? b2[nc] : 0.0f);
        #pragma unroll
        for (int mt = 0; mt < 2; ++mt)
            #pragma unroll
            for (int r = 0; r < 8; ++r)
                sT[mt * 16 + rbase + r][n] = acc[mt][r] + bias;   // cols 126/127 exactly 0
    }
    __syncthreads();

    // ================= phase 5: gather-sum  out[b,o] = sum_k T[b, idx2[b,o,k]] =================
    for (int e = tid; e < BT * NOUT; e += 256) {
        const int r = e / NOUT;
        const int o = e - r * NOUT;
        const size_t b = (size_t)(brow0 + r);
        const int* ip = boi + (b * NOUT + o) * KIDX;
        const float s = sT[r][ip[0]] + sT[r][ip[1]] + sT[r][ip[2]];  // idx==126 hits the 0 column
        out[b * NOUT + o] = s;
    }
}

extern "C" void kernel_launch(void* const* d_in, const int* in_sizes, int n_in,
                              void* d_out, int out_size, void* d_ws, size_t ws_size,
                              hipStream_t stream) {
    const float* x  = (const float*)d_in[0];
    const float* W0 = (const float*)d_in[1];
    const float* b0 = (const float*)d_in[2];
    const float* W1 = (const float*)d_in[3];
    const float* b1 = (const float*)d_in[4];
    const float* W2 = (const float*)d_in[5];
    const float* b2 = (const float*)d_in[6];
    const int*  bii = (const int*)d_in[7];
    const int*  boi = (const int*)d_in[8];
    float* out = (float*)d_out;

    const int B = in_sizes[0] / NIN;        // 1024
    fused_batch_indexed_mlp<<<dim3(B / BT), dim3(256), 0, stream>>>(
        x, W0, b0, W1, b1, W2, b2, bii, boi, out);
}